// BasicNetwork_9096740733136
// MI455X (gfx1250) — compile-verified
//
#include <hip/hip_runtime.h>
#include <hip/hip_bf16.h>
#include <stdint.h>

// Problem constants from the reference:
//   B=2, C=32, H=80, W=240, D = 192/3 = 64, output channels = 2C = 64
#define Bn   2
#define Cn   32
#define Hn   80
#define Wn   240
#define Dn   64
#define CCn  (2 * Cn)   // 64 output channels
#define W4   (Wn / 4)   // 60 float4 per row

// Native clang vector type so __builtin_nontemporal_store accepts it
// (HIP's float4 is a class type and is rejected by the builtin).
typedef float v4f __attribute__((ext_vector_type(4)));

// One workgroup per (b, cc, h) triple. Stage the 240-float input row in LDS
// via CDNA5 async global->LDS DMA, then write Dn masked/shifted rows with
// non-temporal b128 stores (output is write-once, 629 MB >> 192 MB L2).
__global__ __launch_bounds__(256) void costvol_kernel(
    const float* __restrict__ x,
    const float* __restrict__ y,
    float* __restrict__ out)
{
    __shared__ float row[Wn];

    const int tid = threadIdx.x;
    const int bc  = blockIdx.x;                 // flat (b, cc, h)
    const int h   = bc % Hn;
    const int cc  = (bc / Hn) % CCn;
    const int b   = bc / (Hn * CCn);

    const bool isRight = (cc >= Cn);
    const int  cin     = isRight ? (cc - Cn) : cc;
    const float* src   = (isRight ? y : x)
                       + ((size_t)(b * Cn + cin) * Hn + h) * Wn;

    // --- Stage the input row into LDS with async DMA (gfx1250) ---
    // 60 lanes each move one 16-byte chunk: 960 B total.
    if (tid < W4) {
        uint32_t lds_off = (uint32_t)(uintptr_t)(&row[tid * 4]); // low 32 bits of
                                                                 // flat addr == LDS byte addr
        uint64_t gaddr   = (uint64_t)(uintptr_t)(src + tid * 4);
        asm volatile("global_load_async_to_lds_b128 %0, %1, off"
                     :: "v"(lds_off), "v"(gaddr)
                     : "memory");
    }
    asm volatile("s_wait_asynccnt 0" ::: "memory");
    __syncthreads();

    // --- Write Dn output rows for this (b, cc, h) ---
    // out index = (((b*CCn + cc)*Dn + d)*Hn + h)*Wn + w
    const size_t chanStride = (size_t)Dn * Hn * Wn;           // per (b,cc)
    const size_t rowStride  = (size_t)Hn * Wn;                // per d
    float* obase = out + (size_t)(b * CCn + cc) * chanStride + (size_t)h * Wn;

    // Dn*W4 = 64*60 = 3840 float4 stores per block; 15 iterations per thread.
    // Fully unrolled: keeps many NT b128 stores in flight per wave.
#pragma unroll
    for (int k = 0; k < (Dn * W4) / 256; ++k) {
        const int t     = k * 256 + tid;
        const int d     = t / W4;
        const int i4    = t - d * W4;
        const int w0    = i4 * 4;
        const int shift = isRight ? d : 0;

        v4f v;
        v.x = (w0 + 0 >= d) ? row[w0 + 0 - shift] : 0.0f;
        v.y = (w0 + 1 >= d) ? row[w0 + 1 - shift] : 0.0f;
        v.z = (w0 + 2 >= d) ? row[w0 + 2 - shift] : 0.0f;
        v.w = (w0 + 3 >= d) ? row[w0 + 3 - shift] : 0.0f;

        __builtin_nontemporal_store(
            v, reinterpret_cast<v4f*>(obase + (size_t)d * rowStride + w0));
    }
}

extern "C" void kernel_launch(void* const* d_in, const int* in_sizes, int n_in,
                              void* d_out, int out_size, void* d_ws, size_t ws_size,
                              hipStream_t stream) {
    (void)in_sizes; (void)n_in; (void)out_size; (void)d_ws; (void)ws_size;
    const float* x = (const float*)d_in[0];
    const float* y = (const float*)d_in[1];
    float* out = (float*)d_out;

    const int nBlocks = Bn * CCn * Hn;   // 2*64*80 = 10240
    costvol_kernel<<<nBlocks, 256, 0, stream>>>(x, y, out);
}